// LightGCNLayer_50775103373666
// MI455X (gfx1250) — compile-verified
//
#include <hip/hip_runtime.h>
#include <stdint.h>

#define DIM   64
#define CHUNK 256   // edges per block
#define BLOCK 256

__global__ void zero_kernel(float* __restrict__ out, int n) {
    int idx = blockIdx.x * blockDim.x + threadIdx.x;
    int n4 = n >> 2;
    if (idx < n4) {
        reinterpret_cast<float4*>(out)[idx] = make_float4(0.f, 0.f, 0.f, 0.f);
    }
    if (idx == 0) {
        for (int t = (n4 << 2); t < n; ++t) out[t] = 0.f;
    }
}

__global__ __launch_bounds__(BLOCK) void lightgcn_scatter(
    const float* __restrict__ user_emb,
    const float* __restrict__ item_emb,
    const float* __restrict__ edge_norm,
    const int*   __restrict__ u_idx,
    const int*   __restrict__ i_idx,
    float*       __restrict__ agg_users,
    float*       __restrict__ agg_items,
    int num_edges)
{
    __shared__ int   s_u[CHUNK];
    __shared__ int   s_i[CHUNK];
    __shared__ float s_n[CHUNK];

    const int tid        = threadIdx.x;
    const int chunk_base = blockIdx.x * CHUNK;

    // ---- CDNA5 async-to-LDS staging of edge metadata (uses ASYNCcnt) ----
    {
        int e  = chunk_base + tid;
        int ce = (e < num_edges) ? e : (num_edges - 1);   // clamp: keep EXEC all-ones
        int byte_off = ce * 4;
        // Flat pointers to LDS: low 32 bits are the wave-relative LDS byte address.
        uint32_t lds_u = (uint32_t)(uintptr_t)(&s_u[tid]);
        uint32_t lds_i = (uint32_t)(uintptr_t)(&s_i[tid]);
        uint32_t lds_n = (uint32_t)(uintptr_t)(&s_n[tid]);
        asm volatile("global_load_async_to_lds_b32 %0, %1, %2"
                     :: "v"(lds_u), "v"(byte_off), "s"(u_idx) : "memory");
        asm volatile("global_load_async_to_lds_b32 %0, %1, %2"
                     :: "v"(lds_i), "v"(byte_off), "s"(i_idx) : "memory");
        asm volatile("global_load_async_to_lds_b32 %0, %1, %2"
                     :: "v"(lds_n), "v"(byte_off), "s"(edge_norm) : "memory");
        asm volatile("s_wait_asynccnt 0" ::: "memory");
    }
    __syncthreads();

    // 16 groups of 16 lanes; each group processes 16 edges; each lane owns
    // 4 consecutive floats of the 64-dim row -> one global_load_b128 / row / lane,
    // fully coalesced to a single 256B cache line per row.
    const int group = tid >> 4;
    const int lane  = tid & 15;

    #pragma unroll 4
    for (int k = 0; k < 16; ++k) {
        int e_local = (group << 4) + k;
        int e = chunk_base + e_local;
        if (e >= num_edges) break;

        int   u = s_u[e_local];
        int   i = s_i[e_local];
        float n = s_n[e_local];

        const float4 ue = reinterpret_cast<const float4*>(user_emb + (size_t)u * DIM)[lane];
        const float4 ie = reinterpret_cast<const float4*>(item_emb + (size_t)i * DIM)[lane];

        float* it_dst = agg_items + (size_t)i * DIM + (lane << 2);
        float* us_dst = agg_users + (size_t)u * DIM + (lane << 2);

        // Native f32 atomics -> global_atomic_add_f32 (L2-resident RMW)
        unsafeAtomicAdd(it_dst + 0, n * ue.x);
        unsafeAtomicAdd(it_dst + 1, n * ue.y);
        unsafeAtomicAdd(it_dst + 2, n * ue.z);
        unsafeAtomicAdd(it_dst + 3, n * ue.w);
        unsafeAtomicAdd(us_dst + 0, n * ie.x);
        unsafeAtomicAdd(us_dst + 1, n * ie.y);
        unsafeAtomicAdd(us_dst + 2, n * ie.z);
        unsafeAtomicAdd(us_dst + 3, n * ie.w);
    }
}

extern "C" void kernel_launch(void* const* d_in, const int* in_sizes, int n_in,
                              void* d_out, int out_size, void* d_ws, size_t ws_size,
                              hipStream_t stream) {
    const float* user_emb  = (const float*)d_in[0];
    const float* item_emb  = (const float*)d_in[1];
    const float* edge_norm = (const float*)d_in[2];
    const int*   u_idx     = (const int*)d_in[3];
    const int*   i_idx     = (const int*)d_in[4];

    const int num_users = in_sizes[0] / DIM;
    const int num_edges = in_sizes[2];

    float* agg_users = (float*)d_out;
    float* agg_items = agg_users + (size_t)num_users * DIM;

    // d_out is poisoned by the harness: zero it every call (deterministic).
    int n4    = out_size >> 2;
    int zgrid = (n4 + BLOCK - 1) / BLOCK;
    if (zgrid < 1) zgrid = 1;
    zero_kernel<<<zgrid, BLOCK, 0, stream>>>((float*)d_out, out_size);

    int grid = (num_edges + CHUNK - 1) / CHUNK;
    lightgcn_scatter<<<grid, BLOCK, 0, stream>>>(user_emb, item_emb, edge_norm,
                                                 u_idx, i_idx,
                                                 agg_users, agg_items, num_edges);
}